// SwLayer_3221225472359
// MI455X (gfx1250) — compile-verified
//
#include <hip/hip_runtime.h>
#include <math.h>
#include <stdint.h>

typedef float v2f __attribute__((ext_vector_type(2)));
typedef float v8f __attribute__((ext_vector_type(8)));

#define NCLS 100

// ---------------------------------------------------------------------------
// Kernel A: spatial mean over H*W=1024 for each (b,c). One wave32 per row.
// HBM-bound stage: 268 MB of coalesced b128 loads.
// ---------------------------------------------------------------------------
__global__ void mean_kernel(const float* __restrict__ x, float* __restrict__ value)
{
    int wave = threadIdx.x >> 5;
    int lane = threadIdx.x & 31;
    int bc = blockIdx.x * 8 + wave;                 // 512*128/8 = 8192 blocks
    const float4* p = reinterpret_cast<const float4*>(x) + (size_t)bc * 256;
    float s = 0.0f;
#pragma unroll
    for (int it = 0; it < 8; ++it) {
        float4 v = p[lane + 32 * it];
        s += (v.x + v.y) + (v.z + v.w);
    }
#pragma unroll
    for (int off = 16; off > 0; off >>= 1) s += __shfl_xor(s, off, 32);
    if (lane == 0) value[bc] = s * (1.0f / 1024.0f);
}

// ---------------------------------------------------------------------------
// Kernel B: per-class Cholesky (in LDS), logdet, triangular inversion.
// Wt[k][c][i] = (L^-1)[i][c]  (row-major [C][C] per class, i.e. W transposed)
// 128 threads/block, one block per class. Dynamic LDS ~132.6 KB.
// ---------------------------------------------------------------------------
__global__ void chol_kernel(const float* __restrict__ convs,
                            float* __restrict__ Wt,
                            float* __restrict__ logdet)
{
    extern __shared__ float smem[];
    const int C = 128, LDA = 129;                   // pad to avoid bank conflicts
    float* sM  = smem;                              // C*LDA
    float* sW  = smem + C * LDA;                    // C*LDA
    float* red = sW + C * LDA;                      // 128

    int k = blockIdx.x, tid = threadIdx.x;
    const float* src = convs + (size_t)k * C * C;
    for (int idx = tid; idx < C * C; idx += C) {
        int r = idx >> 7, c = idx & 127;
        sM[r * LDA + c] = src[idx];
    }
    __syncthreads();

    // right-looking Cholesky, sM becomes L (lower triangle)
    for (int j = 0; j < C; ++j) {
        if (tid == j) sM[j * LDA + j] = sqrtf(sM[j * LDA + j]);
        __syncthreads();
        float d = sM[j * LDA + j];
        if (tid > j) sM[tid * LDA + j] /= d;
        __syncthreads();
        if (tid > j) {
            float lij = sM[tid * LDA + j];
            for (int m = j + 1; m <= tid; ++m)
                sM[tid * LDA + m] -= lij * sM[m * LDA + j];
        }
        __syncthreads();
    }

    // logdet = 2 * sum(log(diag(L)))
    red[tid] = logf(sM[tid * LDA + tid]);
    __syncthreads();
    for (int s = 64; s > 0; s >>= 1) {
        if (tid < s) red[tid] += red[tid + s];
        __syncthreads();
    }
    if (tid == 0) logdet[k] = 2.0f * red[0];

    // thread j computes column j of L^-1 -> stored as row j of sW
    {
        int j = tid;
        float* wr = sW + j * LDA;
        for (int i = 0; i < j; ++i) wr[i] = 0.0f;
        for (int i = j; i < C; ++i) {
            const float* Li = sM + i * LDA;
            float s0 = 0.f, s1 = 0.f, s2 = 0.f, s3 = 0.f;
            int m = j;
            for (; m + 3 < i; m += 4) {             // 4-way to break FMA chain
                s0 += Li[m]     * wr[m];
                s1 += Li[m + 1] * wr[m + 1];
                s2 += Li[m + 2] * wr[m + 2];
                s3 += Li[m + 3] * wr[m + 3];
            }
            for (; m < i; ++m) s0 += Li[m] * wr[m];
            float num = ((i == j) ? 1.0f : 0.0f) - ((s0 + s1) + (s2 + s3));
            wr[i] = num / Li[i];
        }
    }
    __syncthreads();

    float* dst = Wt + (size_t)k * C * C;
    for (int idx = tid; idx < C * C; idx += C) {
        int r = idx >> 7, c = idx & 127;
        dst[idx] = sW[r * LDA + c];
    }
}

// ---------------------------------------------------------------------------
// Kernel C: probs[b][k] = exp(-0.5*(||Wt^T diff||^2 + C*log2pi + logdet[k]))
// grid = (8 m-strips of 64 rows, 100 classes), 256 threads = 8 waves.
// Wt[k] (64 KB) is staged into LDS via the CDNA5 async-to-LDS engine
// (global_load_async_to_lds_b128 + s_wait_asynccnt), then each wave runs
// 4 WMMA 16x16 tiles reusing one B fragment per k-step.
// ---------------------------------------------------------------------------
__global__ void probs_kernel(const float* __restrict__ value,
                             const float* __restrict__ means,
                             const float* __restrict__ Wt,
                             const float* __restrict__ logdet,
                             float* __restrict__ probs)
{
    const int C = 128, LDB = 132, LDS_A = 132;      // padded strides (bank spread)
    extern __shared__ float smem[];
    float* sB = smem;                               // [128][132]  B tile (Wt[k])
    float* sA = sB + C * LDB;                       // [64][132]   diff tile
    float* sQ = sA + 64 * LDS_A;                    // [64][8]     per-wave row sums

    int k  = blockIdx.y;
    int m0 = blockIdx.x * 64;
    int tid  = threadIdx.x;
    int lane = tid & 31, wave = tid >> 5;
    int half = lane >> 4, l15 = lane & 15;

    // --- async copy Wt[k] (128 rows x 512B) into sB, row stride LDB*4 ---
    {
        const float* WtK = Wt + (size_t)k * C * C;
        unsigned ldsb = (unsigned)(uintptr_t)sB;
        int row0 = wave * 16;                       // 16 rows per wave
#pragma unroll
        for (int r = 0; r < 16; ++r) {
            int row = row0 + r;
            unsigned ldst = ldsb + (unsigned)(row * LDB * 4) + lane * 16;
            unsigned voff = (unsigned)(row * C * 4) + lane * 16;
            asm volatile("global_load_async_to_lds_b128 %0, %1, %2"
                         :: "v"(ldst), "v"(voff), "s"(WtK) : "memory");
        }
    }

    // --- stage diff tile (overlaps with async B copy) ---
    for (int idx = tid; idx < 64 * C; idx += 256) {
        int r = idx >> 7, c = idx & 127;
        sA[r * LDS_A + c] =
            value[(size_t)(m0 + r) * C + c] - means[(size_t)k * C + c];
    }

    asm volatile("s_wait_asynccnt 0x0" ::: "memory");
    __syncthreads();

    int col = wave * 16 + l15;                      // this wave's 16 columns
    v8f acc0 = {0.f,0.f,0.f,0.f,0.f,0.f,0.f,0.f};
    v8f acc1 = acc0, acc2 = acc0, acc3 = acc0;

    for (int kb = 0; kb < C; kb += 4) {
        int kr = kb + half * 2;
        v2f b;
        b.x = sB[kr * LDB + col];
        b.y = sB[(kr + 1) * LDB + col];
        v2f a0, a1, a2, a3;
        a0.x = sA[(l15)      * LDS_A + kr]; a0.y = sA[(l15)      * LDS_A + kr + 1];
        a1.x = sA[(16 + l15) * LDS_A + kr]; a1.y = sA[(16 + l15) * LDS_A + kr + 1];
        a2.x = sA[(32 + l15) * LDS_A + kr]; a2.y = sA[(32 + l15) * LDS_A + kr + 1];
        a3.x = sA[(48 + l15) * LDS_A + kr]; a3.y = sA[(48 + l15) * LDS_A + kr + 1];
        acc0 = __builtin_amdgcn_wmma_f32_16x16x4_f32(false, a0, false, b, (short)0, acc0, false, false);
        acc1 = __builtin_amdgcn_wmma_f32_16x16x4_f32(false, a1, false, b, (short)0, acc1, false, false);
        acc2 = __builtin_amdgcn_wmma_f32_16x16x4_f32(false, a2, false, b, (short)0, acc2, false, false);
        acc3 = __builtin_amdgcn_wmma_f32_16x16x4_f32(false, a3, false, b, (short)0, acc3, false, false);
    }

    // row sums of y^2: reduce each 16x16 tile across its 16 columns
    v8f* accs[4] = {&acc0, &acc1, &acc2, &acc3};
#pragma unroll
    for (int t = 0; t < 4; ++t) {
        float rs[8];
#pragma unroll
        for (int v = 0; v < 8; ++v) { float y = (*accs[t])[v]; rs[v] = y * y; }
#pragma unroll
        for (int off = 1; off < 16; off <<= 1) {
#pragma unroll
            for (int v = 0; v < 8; ++v) rs[v] += __shfl_xor(rs[v], off, 32);
        }
        if (l15 == 0) {
#pragma unroll
            for (int v = 0; v < 8; ++v)
                sQ[(t * 16 + v + half * 8) * 8 + wave] = rs[v];
        }
    }
    __syncthreads();

    if (tid < 64) {
        float q = 0.f;
#pragma unroll
        for (int w = 0; w < 8; ++w) q += sQ[tid * 8 + w];
        const float log2pi = 1.8378770664093453f;
        float lp = -0.5f * (q + 128.0f * log2pi + logdet[k]);
        probs[(size_t)(m0 + tid) * NCLS + k] = expf(lp);
    }
}

// ---------------------------------------------------------------------------
// Kernel D: Out = act(A[M,K] @ W[N,K]^T + bias), act = leaky-relu(0.01) or id.
// grid = (M/16, ceil(N/128)), 256 threads = 8 waves, wave -> one 16x16 tile.
// ---------------------------------------------------------------------------
__global__ void mlp_gemm(const float* __restrict__ A, const float* __restrict__ W,
                         const float* __restrict__ bias, float* __restrict__ Out,
                         int M, int N, int K, int act)
{
    extern __shared__ float sA[];
    int lda = K + 4;
    int m0 = blockIdx.x * 16;
    int tid = threadIdx.x;

    for (int idx = tid; idx < 16 * K; idx += 256) {
        int r = idx / K, c = idx - r * K;
        sA[r * lda + c] = A[(size_t)(m0 + r) * K + c];
    }
    __syncthreads();

    int lane = tid & 31, wave = tid >> 5;
    int half = lane >> 4, l15 = lane & 15;
    int col = blockIdx.y * 128 + wave * 16 + l15;
    bool colok = col < N;
    const float* Wc = W + (size_t)(colok ? col : 0) * K;  // clamp, mask at store

    v8f acc = {0.f, 0.f, 0.f, 0.f, 0.f, 0.f, 0.f, 0.f};
    for (int kb = 0; kb < K; kb += 4) {
        int kr = kb + half * 2;
        v2f a, b;
        a.x = sA[l15 * lda + kr];
        a.y = sA[l15 * lda + kr + 1];
        b.x = Wc[kr];
        b.y = Wc[kr + 1];
        acc = __builtin_amdgcn_wmma_f32_16x16x4_f32(false, a, false, b,
                                                    (short)0, acc, false, false);
    }

    float bv = colok ? bias[col] : 0.0f;
#pragma unroll
    for (int v = 0; v < 8; ++v) {
        float r = acc[v] + bv;
        if (act) r = (r >= 0.0f) ? r : 0.01f * r;
        int row = m0 + v + half * 8;
        if (colok) Out[(size_t)row * N + col] = r;
    }
}

// ---------------------------------------------------------------------------
extern "C" void kernel_launch(void* const* d_in, const int* in_sizes, int n_in,
                              void* d_out, int out_size, void* d_ws, size_t ws_size,
                              hipStream_t stream)
{
    const float* x     = (const float*)d_in[0];
    const float* means = (const float*)d_in[1];
    const float* convs = (const float*)d_in[2];
    const float* w1 = (const float*)d_in[3];
    const float* b1 = (const float*)d_in[4];
    const float* w2 = (const float*)d_in[5];
    const float* b2 = (const float*)d_in[6];
    const float* w3 = (const float*)d_in[7];
    const float* b3 = (const float*)d_in[8];
    float* out = (float*)d_out;

    float* ws = (float*)d_ws;
    float* value  = ws;  ws += 512 * 128;
    float* Wt     = ws;  ws += (size_t)100 * 128 * 128;
    float* logdet = ws;  ws += 128;
    float* probs  = ws;  ws += 512 * NCLS;
    float* H1     = ws;  ws += 512 * 512;
    float* H2     = ws;  ws += 512 * 512;

    // large dynamic LDS (CDNA5: up to 320 KB per workgroup)
    (void)hipFuncSetAttribute((const void*)chol_kernel,
                              hipFuncAttributeMaxDynamicSharedMemorySize,
                              140 * 1024);
    (void)hipFuncSetAttribute((const void*)probs_kernel,
                              hipFuncAttributeMaxDynamicSharedMemorySize,
                              112 * 1024);

    mean_kernel<<<8192, 256, 0, stream>>>(x, value);

    size_t cholLds = (size_t)(2 * 128 * 129 + 128) * sizeof(float);
    chol_kernel<<<100, 128, cholLds, stream>>>(convs, Wt, logdet);

    size_t probsLds = (size_t)(128 * 132 + 64 * 132 + 64 * 8) * sizeof(float);
    probs_kernel<<<dim3(8, 100), 256, probsLds, stream>>>(value, means, Wt,
                                                          logdet, probs);

    mlp_gemm<<<dim3(32, 4), 256, (size_t)16 * (100 + 4) * 4, stream>>>(
        probs, w1, b1, H1, 512, 512, 100, 1);
    mlp_gemm<<<dim3(32, 4), 256, (size_t)16 * (512 + 4) * 4, stream>>>(
        H1, w2, b2, H2, 512, 512, 512, 1);
    mlp_gemm<<<dim3(32, 1), 256, (size_t)16 * (512 + 4) * 4, stream>>>(
        H2, w3, b3, out, 512, 100, 512, 0);
}